// BatchBlur_nopad_4105988735746
// MI455X (gfx1250) — compile-verified
//
#include <hip/hip_runtime.h>
#include <stdint.h>

// ---- problem constants (from the reference) ----
#define LK    19           // filter size
#define PAD   9            // L/2
#define CIN   3
#define HIN   256
#define WIN   256
#define HO    238          // 256 - 18
#define WO    238
#define K2    361          // 19*19
#define PLANE (HIN * WIN)  // 65536 elements per (b,k) kernel plane

// ---- tiling ----
#define TW    32           // output tile width  (1 wave row, fully coalesced)
#define THT   8            // output tile height (256 threads/block)
#define IW    (TW + LK - 1)   // 50 input cols staged
#define IH    (THT + LK - 1)  // 26 input rows staged
#define LDS_CH (IW * IH)      // 1300 floats per channel

typedef __attribute__((ext_vector_type(4))) unsigned int v4u;
typedef __attribute__((ext_vector_type(8))) int          v8i;
typedef __attribute__((ext_vector_type(4))) int          v4i;

__global__ __launch_bounds__(256) void
batchblur_nopad_kernel(const float* __restrict__ inp,
                       const float* __restrict__ ker,
                       float* __restrict__ out)
{
    __shared__ float tile[CIN * LDS_CH];   // 15,600 B of the 320 KB WGP LDS

    const int tid = threadIdx.x;
    const int b   = blockIdx.z;

    // Clamp edge tiles inward: overlapping blocks recompute identical values.
    int w0 = blockIdx.x * TW;  if (w0 > WO - TW)  w0 = WO - TW;
    int h0 = blockIdx.y * THT; if (h0 > HO - THT) h0 = HO - THT;

    // ---------------------------------------------------------------
    // Stage the 3-channel input tile (50 x 26 x 3 f32) into LDS via TDM.
    // ---------------------------------------------------------------
#if __has_builtin(__builtin_amdgcn_tensor_load_to_lds) && __has_builtin(__builtin_amdgcn_s_wait_tensorcnt)
    if (tid < 32) {   // wave 0 issues a single TDM descriptor (EXEC ignored by TDM)
        const uint64_t gaddr =
            (uint64_t)(uintptr_t)(inp + ((size_t)b * CIN * HIN + (size_t)h0) * WIN + w0);
        const uint32_t laddr = (uint32_t)(uintptr_t)(&tile[0]);  // LDS byte offset

        // ---- D# group 0 (ISA 8.3): count=1, lds_addr, global_addr, type=2 ----
        v4u g0;
        g0[0] = 1u;                                        // count=1, user mode, no gather
        g0[1] = laddr;                                     // lds_addr [63:32]
        g0[2] = (uint32_t)(gaddr & 0xFFFFFFFFu);           // global_addr[31:0]
        g0[3] = (uint32_t)((gaddr >> 32) & 0x1FFFFFFu)     // global_addr[56:32]
              | (2u << 30);                                // type=2 ("image")

        // ---- D# group 1 (ISA 8.4) ----
        // x = column (50 of 256, stride 1), y = row (26 of 256, stride 256),
        // z = channel (3, stride 65536). data_size=2 -> 4 bytes.
        v8i g1;
        g1[0] = (int)(2u << 16);                                   // data_size=4B, mask=0
        g1[1] = (int)((uint32_t)WIN << 16);                        // tensor_dim0[15:0]
        g1[2] = (int)(((uint32_t)WIN >> 16) | ((uint32_t)HIN << 16)); // dim0[31:16] | dim1[15:0]
        g1[3] = (int)(((uint32_t)HIN >> 16) | ((uint32_t)IW << 16));  // dim1[31:16] | tile_dim0
        g1[4] = (int)((uint32_t)IH | ((uint32_t)CIN << 16));       // tile_dim1 | tile_dim2
        g1[5] = (int)(uint32_t)WIN;                                // tensor_dim0_stride = 256
        g1[6] = (int)(((uint32_t)PLANE & 0xFFFFu) << 16);          // stride0[47:32]=0 | stride1[15:0]
        g1[7] = (int)((uint32_t)PLANE >> 16);                      // stride1[47:16] (65536>>16 = 1)

        // ---- D# group 2: tensor_dim2 = 3 channels; rest unused (3-D tile) ----
        v4i g2; g2[0] = CIN; g2[1] = 0; g2[2] = 0; g2[3] = 0;
        v4i g3 = {0, 0, 0, 0};
        // 6-arg toolchain form: extra int32x8 group (reserved here) zero-filled.
        v8i g4 = {0, 0, 0, 0, 0, 0, 0, 0};

        __builtin_amdgcn_tensor_load_to_lds(g0, g1, g2, g3, g4, /*cpol=*/0);
        __builtin_amdgcn_s_wait_tensorcnt(0);   // TENSORcnt == 0: tile is in LDS
    }
    asm volatile("" ::: "memory");
#else
    // Fallback: cooperative vector-unit staging of the same SoA tile.
    for (int idx = tid; idx < CIN * LDS_CH; idx += 256) {
        const int c = idx / LDS_CH;
        const int r = idx - c * LDS_CH;
        const int y = r / IW;
        const int x = r - y * IW;
        tile[idx] = inp[(((size_t)b * CIN + c) * HIN + (h0 + y)) * WIN + (w0 + x)];
    }
#endif
    __syncthreads();

    // ---------------------------------------------------------------
    // Each lane: one output pixel, 3 channel accumulators.
    // ---------------------------------------------------------------
    const int tx = tid & (TW - 1);
    const int ty = tid >> 5;
    const int h  = h0 + ty;
    const int w  = w0 + tx;

    // Base pointer for this lane's kernel weights: ker[b, k, h+9, w+9],
    // per-k stride = 64K elements (256 KB). Lanes are consecutive in w ->
    // every kernel load is a fully-coalesced 128 B wave transaction.
    const float* kp = ker + (size_t)b * K2 * PLANE
                          + (size_t)(h + PAD) * WIN + (w + PAD);
    const float* t0 = &tile[ty * IW + tx];

    float a0 = 0.f, a1 = 0.f, a2 = 0.f;

    for (int i = 0; i < LK; ++i) {
        const float* kr = kp + (size_t)(i * LK) * PLANE;  // one 64-bit base per tap row
        const float* tr = t0 + i * IW;
#pragma unroll
        for (int j = 0; j < LK; ++j) {
            // Single-use stream: non-temporal so the 378 MB kernel tensor
            // does not evict the L2-resident input.
            const float kv = __builtin_nontemporal_load(kr + (size_t)j * PLANE);
            const float x0 = tr[j];
            const float x1 = tr[j + LDS_CH];
            const float x2 = tr[j + 2 * LDS_CH];
            a0 = fmaf(kv, x0, a0);
            a1 = fmaf(kv, x1, a1);
            a2 = fmaf(kv, x2, a2);
        }
    }

    float* op = out + (((size_t)b * CIN) * HO + h) * WO + w;
    __builtin_nontemporal_store(a0, op);
    __builtin_nontemporal_store(a1, op + (size_t)HO * WO);
    __builtin_nontemporal_store(a2, op + 2 * (size_t)HO * WO);
}

extern "C" void kernel_launch(void* const* d_in, const int* in_sizes, int n_in,
                              void* d_out, int out_size, void* d_ws, size_t ws_size,
                              hipStream_t stream) {
    const float* inp = (const float*)d_in[0];   // (4, 3, 256, 256) f32
    const float* ker = (const float*)d_in[1];   // (4, 361, 256, 256) f32
    float*       out = (float*)d_out;           // (4, 3, 238, 238) f32

    dim3 grid((WO + TW - 1) / TW,    // 8
              (HO + THT - 1) / THT,  // 30
              4);                    // batch
    batchblur_nopad_kernel<<<grid, dim3(256), 0, stream>>>(inp, ker, out);
}